// WaveCell_8890582303449
// MI455X (gfx1250) — compile-verified
//
#include <hip/hip_runtime.h>
#include <math.h>

#define NXg 200
#define NYg 200
#define NCELL (NXg*NYg)
#define NSTEP 300
#define BATCH 8
#define NCHUNK (NCELL/4)           // 10000 4-wide row-aligned chunks
#define SRC_CH 1525                // cell 6100  = (30,100), elem .x
#define PB0_CH 8512                // cell 34050 = (170,50), elem .z
#define PB1_CH 8525                // cell 34100 = (170,100), elem .x
#define PB2_CH 8537                // cell 34150 = (170,150), elem .z
#define UROWS 403                  // 200+200 field rows + 3 shared halo rows
#define UFLOATS (UROWS*NYg)        // 80600 floats = 322,400 B
#define HH (1.005f*1.005f)         // H^2, H = DT*2.01

typedef float v2f __attribute__((ext_vector_type(2)));
typedef float v8f __attribute__((ext_vector_type(8)));

// ---------------------------------------------------------------------------
// Kernel 1: per-cell coefficients into d_ws.
//   A    = 1/(4+b)   (DT=0.5: DT^-2 = 4, 0.5*b/DT = b)
//   A2   = 8/(4+b)   (= A*2/DT^2) ;  A*Bm = A2 - 1 (exact identity)
//   R    = A*c2/H^2
// coef[c] = { A2, R }
// ---------------------------------------------------------------------------
__global__ __launch_bounds__(256)
void coef_kernel(const float* __restrict__ rho, float2* __restrict__ coef)
{
    int c = blockIdx.x * blockDim.x + threadIdx.x;
    if (c >= NCELL) return;
    int i = c / NYg, j = c - i * NYg;

    float cen = rho[c];
    float up  = (i > 0)       ? rho[c - NYg] : 0.f;
    float dn  = (i < NXg - 1) ? rho[c + NYg] : 0.f;
    float wv  = (j > 0)       ? rho[c - 1]   : 0.f;
    float ev  = (j < NYg - 1) ? rho[c + 1]   : 0.f;
    float lp  = 0.5f * cen + 0.125f * (up + dn + wv + ev);

    float t_eta = tanhf(50.f);
    float rho_p = (t_eta + tanhf(100.f * (lp - 0.5f))) / (t_eta + tanhf(50.f));
    float cc    = 1.0f - 0.1f * rho_p;
    float c2    = cc * cc;

    double di = 0.0, dj = 0.0;
    if (i <= 20) di = (20 - i) / 20.0; else if (i >= 179) di = (i - 179) / 20.0;
    if (j <= 20) dj = (20 - j) / 20.0; else if (j >= 179) dj = (j - 179) / 20.0;
    double bx = 3.0 * di * di * di * di;
    double by = 3.0 * dj * dj * dj * dj;
    float  b  = (float)sqrt(bx * bx + by * by);

    float Ainv = 4.0f + b;
    coef[c] = make_float2(8.0f / Ainv, c2 / (Ainv * HH));
}

// ---------------------------------------------------------------------------
// Kernel 2: 300-step evolution, fields LDS-resident, one WG per batch item.
// Ping-pong buffers share the middle halo row: rows [0]=halo, [1..200]=buf0,
// [201]=shared halo, [202..401]=buf1, [402]=halo -> 322,400 B. Probe history
// P[t][0..3] (col 3 permanently zero, feeds zero rows of the WMMA fragments)
// -> 4,800 B. Total 327,200 B <= 320 KB WG limit.
// Inner loop: 4-wide row-aligned chunks, b128 LDS traffic, b128 coef loads.
// Final: I_k = diag(P^T P) via 75 chained V_WMMA_F32_16X16X4_F32 on wave 0.
// ---------------------------------------------------------------------------
__global__ __launch_bounds__(1024)
void wave_kernel(const float* __restrict__ x,
                 const float2* __restrict__ coef,
                 float* __restrict__ out)
{
    __shared__ __align__(16) float Ubase[UFLOATS];
    __shared__ __align__(16) float P[NSTEP * 4];

    const int tid = threadIdx.x;
    const int b   = blockIdx.x;
    const float* __restrict__ cfp = (const float*)coef;

    for (int k = tid; k < UFLOATS; k += 1024) Ubase[k] = 0.f;
    for (int k = tid; k < NSTEP * 4; k += 1024) P[k] = 0.f;
    __syncthreads();

    int cur = 0;
    for (int t = 0; t < NSTEP; ++t) {
        const float xt = x[b * NSTEP + t];
        const float* __restrict__ u1 = Ubase + NYg + cur       * 40200;
        float* __restrict__       u2 = Ubase + NYg + (1 - cur) * 40200;

        for (int c4 = tid; c4 < NCHUNK; c4 += 1024) {
            const int i   = c4 / 50;
            const int j0  = (c4 - i * 50) * 4;
            const int idx = i * NYg + j0;           // == first cell index

            const float4 yc = *(const float4*)(u1 + idx);
            const float4 yn = *(const float4*)(u1 + idx - NYg);
            const float4 ys = *(const float4*)(u1 + idx + NYg);
            const float  yw = (j0 != 0)       ? u1[idx - 1] : 0.f;
            const float  ye = (j0 != NYg - 4) ? u1[idx + 4] : 0.f;
            const float4 y2 = *(const float4*)(u2 + idx);
            const float4 ca = *(const float4*)(cfp + 2 * idx);     // {A2,R}x2
            const float4 cb = *(const float4*)(cfp + 2 * idx + 4);

            float4 yo;
            {
                const float lap0 = yn.x + ys.x + yw   + yc.y - 4.f * yc.x;
                const float lap1 = yn.y + ys.y + yc.x + yc.z - 4.f * yc.y;
                const float lap2 = yn.z + ys.z + yc.y + yc.w - 4.f * yc.z;
                const float lap3 = yn.w + ys.w + yc.z + ye   - 4.f * yc.w;
                yo.x = ca.x * yc.x - (ca.x - 1.f) * y2.x + ca.y * lap0;
                yo.y = ca.z * yc.y - (ca.z - 1.f) * y2.y + ca.w * lap1;
                yo.z = cb.x * yc.z - (cb.x - 1.f) * y2.z + cb.y * lap2;
                yo.w = cb.z * yc.w - (cb.z - 1.f) * y2.w + cb.w * lap3;
            }
            if (c4 == SRC_CH) yo.x += xt;
            *(float4*)(u2 + idx) = yo;
            if (c4 == PB0_CH) P[t * 4 + 0] = yo.z;
            if (c4 == PB1_CH) P[t * 4 + 1] = yo.x;
            if (c4 == PB2_CH) P[t * 4 + 2] = yo.z;
        }
        cur ^= 1;
        __syncthreads();
    }

    // ---- I_k = sum_t P[t][k]^2 = diag(P^T P) via f32 WMMA on wave 0 -------
    if (tid < 32) {                              // full wave, EXEC all ones
        const int m     = tid & 15;
        const int mc    = (m < 3) ? m : 3;       // rows m>=3 read zero col 3
        const int khalf = (tid >> 4) << 1;       // K pair {0,1} / {2,3}
        v8f acc = {};
        for (int q = 0; q < NSTEP / 4; ++q) {
            const int t0 = 4 * q + khalf;
            v2f a;                               // Gram product: A == B frag
            a.x = P[t0 * 4 + mc];
            a.y = P[t0 * 4 + 4 + mc];
            acc = __builtin_amdgcn_wmma_f32_16x16x4_f32(
                      false, a, false, a, (short)0, acc, false, false);
        }
        // diag (m,m) for m<8 lives in VGPR m, lane m
        if      (tid == 0) P[0] = acc[0];
        else if (tid == 1) P[1] = acc[1];
        else if (tid == 2) P[2] = acc[2];
    }
    __syncthreads();

    if (tid < 3) {
        const float s = P[0] + P[1] + P[2];
        out[b * 3 + tid] = P[tid] / s;
    }
}

// ---------------------------------------------------------------------------
extern "C" void kernel_launch(void* const* d_in, const int* in_sizes, int n_in,
                              void* d_out, int out_size, void* d_ws, size_t ws_size,
                              hipStream_t stream)
{
    const float* x   = (const float*)d_in[0];   // (8, 300) f32
    const float* rho = (const float*)d_in[1];   // (200, 200) f32
    float2* coef = (float2*)d_ws;               // 320 KB scratch

    coef_kernel<<<(NCELL + 255) / 256, 256, 0, stream>>>(rho, coef);
    wave_kernel<<<BATCH, 1024, 0, stream>>>(x, coef, (float*)d_out);
}